// StruGNN_4956392259830
// MI455X (gfx1250) — compile-verified
//
#include <hip/hip_runtime.h>

#define DFEAT 256
#define TPB 256

typedef __attribute__((ext_vector_type(2))) float v2f;
typedef __attribute__((ext_vector_type(8))) float v8f;

// ---------------- degree / dinv ----------------

__global__ void strugnn_deg_init(float* __restrict__ deg, int n) {
    int i = blockIdx.x * blockDim.x + threadIdx.x;
    if (i < n) deg[i] = 1.0f;  // self loop contributes 1 to every degree
}

__global__ void strugnn_deg_edges(const int* __restrict__ edges, float* __restrict__ deg, int ne) {
    int i = blockIdx.x * blockDim.x + threadIdx.x;
    if (i < ne) {
        int u = edges[2 * i];
        int v = edges[2 * i + 1];
        atomicAdd(&deg[u], 1.0f);   // exact: integer-valued f32 sums
        atomicAdd(&deg[v], 1.0f);
    }
}

__global__ void strugnn_dinv(const float* __restrict__ deg, float* __restrict__ dinv, int n) {
    int i = blockIdx.x * blockDim.x + threadIdx.x;
    if (i < n) dinv[i] = rsqrtf(fmaxf(deg[i], 1.0f));   // v_rsq_f32, matches jax.lax.rsqrt tol
}

// ---------------- GCN layer: self-loop term (init of output) ----------------

template <bool RELU>
__global__ void strugnn_self_init(const float* __restrict__ hin, const float* __restrict__ dinv,
                                  float* __restrict__ hout, long long nv4) {
    long long i = (long long)blockIdx.x * blockDim.x + threadIdx.x;
    if (i >= nv4) return;
    int row = (int)(i >> 6);                     // 64 float4 per 256-wide row
    float4 x = reinterpret_cast<const float4*>(hin)[i];
    if (RELU) {
        x.x = fmaxf(x.x, 0.f); x.y = fmaxf(x.y, 0.f);
        x.z = fmaxf(x.z, 0.f); x.w = fmaxf(x.w, 0.f);
    }
    float di = dinv[row];
    float s = di * di;                           // self-loop weight dinv[i]^2
    x.x *= s; x.y *= s; x.z *= s; x.w *= s;
    reinterpret_cast<float4*>(hout)[i] = x;
}

// ---------------- GCN layer: edge scatter-add (both directions) ----------------

template <bool RELU>
__global__ __launch_bounds__(TPB)
void strugnn_edge_pass(const float* __restrict__ hin, const int* __restrict__ edges,
                       const float* __restrict__ dinv, float* __restrict__ hout,
                       long long W /* = E*64 */) {
    const long long stride = (long long)gridDim.x * blockDim.x;
    for (long long i = (long long)blockIdx.x * blockDim.x + threadIdx.x; i < W; i += stride) {
        const long long e = i >> 6;
        const int q = (int)(i & 63);             // which float4 of the 256-wide row
        const int u = edges[2 * e];
        const int v = edges[2 * e + 1];

        // Prefetch next iteration's gathered rows (global_prefetch_b8) — hides
        // the random-row L2 latency behind this iteration's atomics.
        const long long nx = i + stride;
        if (nx < W) {
            const long long e2 = nx >> 6;
            const int q2 = (int)(nx & 63);
            const int u2 = edges[2 * e2];
            const int v2 = edges[2 * e2 + 1];
            __builtin_prefetch(&hin[(size_t)u2 * DFEAT + q2 * 4], 0, 0);
            __builtin_prefetch(&hin[(size_t)v2 * DFEAT + q2 * 4], 0, 0);
        }

        const float w = dinv[u] * dinv[v];       // symmetric norm coeff, both directions
        float4 xu = *reinterpret_cast<const float4*>(&hin[(size_t)u * DFEAT + q * 4]);
        float4 xv = *reinterpret_cast<const float4*>(&hin[(size_t)v * DFEAT + q * 4]);
        if (RELU) {
            xu.x = fmaxf(xu.x, 0.f); xu.y = fmaxf(xu.y, 0.f);
            xu.z = fmaxf(xu.z, 0.f); xu.w = fmaxf(xu.w, 0.f);
            xv.x = fmaxf(xv.x, 0.f); xv.y = fmaxf(xv.y, 0.f);
            xv.z = fmaxf(xv.z, 0.f); xv.w = fmaxf(xv.w, 0.f);
        }
        float* pv = &hout[(size_t)v * DFEAT + q * 4];
        float* pu = &hout[(size_t)u * DFEAT + q * 4];
        atomicAdd(pv + 0, xu.x * w); atomicAdd(pv + 1, xu.y * w);
        atomicAdd(pv + 2, xu.z * w); atomicAdd(pv + 3, xu.w * w);
        atomicAdd(pu + 0, xv.x * w); atomicAdd(pu + 1, xv.y * w);
        atomicAdd(pu + 2, xv.z * w); atomicAdd(pu + 3, xv.w * w);
    }
}

// ---------------- L2 row-normalize via WMMA Gram diagonal ----------------
// One wave32 per 16-row tile. A (16x4 f32, M=lane-mod-16) and B (4x16, N=lane-mod-16)
// have identical VGPR layouts when B = A^T, so passing the same registers computes
// C += X_chunk * X_chunk^T; the diagonal of the accumulated 16x16 Gram is the
// per-row sum of squares over all 256 columns.
//
// Tail handling is branch-free: out-of-range rows are CLAMPED to the last valid
// row (reads valid memory; only pollutes Gram-diagonal slots that are never used),
// so the k-loop loads are unguarded and can pipeline ahead of the WMMA chain.

__global__ __launch_bounds__(TPB)
void strugnn_l2norm_wmma(float* __restrict__ h, int nrows) {
    __shared__ float ldsC[8][256];    // per-wave dump of the 8 accumulator VGPRs
    __shared__ float ldsInv[8][16];   // per-wave 16 row scales

    const int wave = threadIdx.x >> 5;
    const int lane = threadIdx.x & 31;
    const int tile = blockIdx.x * 8 + wave;
    const int rbase = tile * 16;
    const bool active = (rbase < nrows);          // wave-uniform

    const int row = rbase + (lane & 15);
    const int rowc = min(row, nrows - 1);         // clamp: tail-safe, branch-free
    // lanes 0-15 feed K pairs (k0, k0+1); lanes 16-31 feed (k0+2, k0+3)
    const float* rp = h + (size_t)rowc * DFEAT + ((lane >> 4) << 1);

    v8f c = {};
    if (active) {
#pragma unroll 8
        for (int k0 = 0; k0 < DFEAT; k0 += 4) {
            v2f a = *reinterpret_cast<const v2f*>(rp + k0);
            // (neg_a, A, neg_b, B, c_mod, C, reuse_a, reuse_b)
            c = __builtin_amdgcn_wmma_f32_16x16x4_f32(false, a, false, a,
                                                      (short)0, c, false, false);
        }
#pragma unroll
        for (int r = 0; r < 8; ++r) ldsC[wave][r * 32 + lane] = c[r];
    }
    __syncthreads();

    if (active && lane < 16) {
        int m = lane;
        // C layout: VGPR r, lanes 0-15 -> (M=r, N=lane); lanes 16-31 -> (M=8+r, N=lane-16)
        float ss = (m < 8) ? ldsC[wave][m * 32 + m]
                           : ldsC[wave][(m - 8) * 32 + (m + 16)];
        float nrm = sqrtf(ss);
        ldsInv[wave][m] = 1.0f / fmaxf(nrm, 1e-12f);
    }
    __syncthreads();

    if (active) {
        for (int idx = lane; idx < 16 * (DFEAT / 4); idx += 32) {
            int r = idx >> 6;                 // row within tile
            int rowg = rbase + r;
            if (rowg < nrows) {
                float4* p = reinterpret_cast<float4*>(h + (size_t)rowg * DFEAT) + (idx & 63);
                float4 vv = *p;
                float s = ldsInv[wave][r];
                vv.x *= s; vv.y *= s; vv.z *= s; vv.w *= s;
                *p = vv;
            }
        }
    }
}

// ---------------- seed gather ----------------

__global__ void strugnn_gather(const float* __restrict__ ent, const int* __restrict__ seeds,
                               float* __restrict__ out, int ns) {
    int i = blockIdx.x * blockDim.x + threadIdx.x;
    if (i >= ns * 64) return;
    int s = i >> 6;
    int q = i & 63;
    int row = seeds[s];
    reinterpret_cast<float4*>(out)[(size_t)s * 64 + q] =
        reinterpret_cast<const float4*>(ent)[(size_t)row * 64 + q];
}

// ---------------- launcher ----------------

extern "C" void kernel_launch(void* const* d_in, const int* in_sizes, int n_in,
                              void* d_out, int out_size, void* d_ws, size_t ws_size,
                              hipStream_t stream) {
    (void)n_in; (void)out_size; (void)ws_size;
    const int D = DFEAT;
    const int N = in_sizes[0] / D;     // 100000
    const int E = in_sizes[2] / 2;     // 400000
    const int S = in_sizes[4];         // 10000

    const float* feats[2] = {(const float*)d_in[0], (const float*)d_in[1]};
    const int*   edges[2] = {(const int*)d_in[2],   (const int*)d_in[3]};
    const int*   seeds[2] = {(const int*)d_in[4],   (const int*)d_in[5]};
    // d_in[6], d_in[7] (triples) are unused by the reference.

    float* out = (float*)d_out;
    // workspace: h1 [N*D] | deg [N] | dinv [N]   (~103 MB, reused per side)
    float* h1   = (float*)d_ws;
    float* deg  = h1 + (size_t)N * D;
    float* dinv = deg + N;

    const long long nv4 = (long long)N * (D / 4);
    const long long W   = (long long)E * 64;
    const int gN   = (N + TPB - 1) / TPB;
    const int gE   = (E + TPB - 1) / TPB;
    const int gV4  = (int)((nv4 + TPB - 1) / TPB);
    const int gSeed = (int)(((long long)S * 64 + TPB - 1) / TPB);
    const int ntiles = (N + 15) / 16;
    const int gNorm = (ntiles + 7) / 8;
    const int gEdge = 4096;            // persistent grid: ~1M threads, ~24 iters over E*64

    for (int side = 0; side < 2; ++side) {
        float* ent = out + (size_t)2 * S * D + (size_t)side * N * D;
        float* sd  = out + (size_t)side * S * D;

        strugnn_deg_init<<<gN, TPB, 0, stream>>>(deg, N);
        strugnn_deg_edges<<<gE, TPB, 0, stream>>>(edges[side], deg, E);
        strugnn_dinv<<<gN, TPB, 0, stream>>>(deg, dinv, N);

        // layer 1: feats -> h1 (self term init, then edge scatter-adds)
        strugnn_self_init<false><<<gV4, TPB, 0, stream>>>(feats[side], dinv, h1, nv4);
        strugnn_edge_pass<false><<<gEdge, TPB, 0, stream>>>(feats[side], edges[side], dinv, h1, W);

        // layer 2 (ReLU fused into reads of h1): h1 -> ent (in d_out)
        strugnn_self_init<true><<<gV4, TPB, 0, stream>>>(h1, dinv, ent, nv4);
        strugnn_edge_pass<true><<<gEdge, TPB, 0, stream>>>(h1, edges[side], dinv, ent, W);

        // row L2 normalize in place (WMMA Gram diagonal), then seed gather
        strugnn_l2norm_wmma<<<gNorm, TPB, 0, stream>>>(ent, N);
        strugnn_gather<<<gSeed, TPB, 0, stream>>>(ent, seeds[side], sd, S);
    }
}